// InvariantPointAttention_10677288697897
// MI455X (gfx1250) — compile-verified
//
#include <hip/hip_runtime.h>
#include <hip/hip_bf16.h>
#include <math.h>

// ---------------------------------------------------------------------------
// Invariant Point Attention on MI455X (gfx1250).
// f32 WMMA everywhere + Tensor Data Mover (TDM) staging for the z tensor.
// B=2, N=512, C_S=384, C_Z=128, H=12, C=16, PQ=4, PV=8, OUT_IN=2112
// ---------------------------------------------------------------------------

#define BB    2
#define NN    512
#define CS    384
#define CZ    128
#define HH    12
#define CC    16
#define PQn   4
#define PVn   8
#define OUTIN 2112
#define WC_   0.23570226039551584f   /* sqrt(2/(9*PQ)) */
#define WL_   0.57735026918962576f   /* sqrt(1/3) */

typedef float v2f  __attribute__((ext_vector_type(2)));
typedef float v8f  __attribute__((ext_vector_type(8)));
typedef unsigned int u32x4 __attribute__((ext_vector_type(4)));
typedef int   i32x4 __attribute__((ext_vector_type(4)));
typedef int   i32x8 __attribute__((ext_vector_type(8)));

static __device__ __forceinline__ v8f wmma4(v2f a, v2f b, v8f c) {
  // V_WMMA_F32_16X16X4_F32 : D(16x16,f32) = A(16x4,f32) * B(4x16,f32) + C
  return __builtin_amdgcn_wmma_f32_16x16x4_f32(false, a, false, b, (short)0, c,
                                               false, false);
}

// ---------------------------------------------------------------------------
// TDM: DMA a 2-D tile of f32 (128 cols x `rows` rows, row stride 128 elems)
// from global memory into LDS. D# bitfields per CDNA5 ISA ch.8:
//   group0: count=1 | lds_addr | global_addr | type=2
//   group1: data_size=4B, tensor_dim0=128, tensor_dim1=512, tile_dim0=128,
//           tile_dim1=rows, tensor_dim0_stride=128
// Tracked by TENSORcnt (in-order per wave) -> s_wait_tensorcnt.
// This toolchain uses the 6-arg builtin form:
//   (u32x4 g0, i32x8 g1, i32x4 g2, i32x4 g3, i32x8 extra, i32 cpol)
// ---------------------------------------------------------------------------
static __device__ __forceinline__ void tdm_load_z_tile(unsigned lds_off,
                                                       const float* gptr,
                                                       int rows) {
  unsigned long long ga = (unsigned long long)(const void*)gptr;
  u32x4 g0;
  g0[0] = 1u;                                          // count=1 (user D#)
  g0[1] = lds_off;                                     // LDS byte address
  g0[2] = (unsigned)(ga & 0xffffffffu);                // global_addr[31:0]
  g0[3] = (unsigned)((ga >> 32) & 0x01ffffffu)         // global_addr[56:32]
        | (2u << 30);                                  // type = 2 ("image")
  i32x8 g1;
  g1[0] = (int)(2u << 16);          // workgroup_mask=0, data_size=2 (4 bytes)
  g1[1] = (int)(128u << 16);        // tensor_dim0[15:0] = 128
  g1[2] = (int)(512u << 16);        // tensor_dim0 hi = 0, tensor_dim1 lo = 512
  g1[3] = (int)(128u << 16);        // tensor_dim1 hi = 0, tile_dim0 = 128
  g1[4] = rows & 0xffff;            // tile_dim1 = rows, tile_dim2 = 0
  g1[5] = 128;                      // tensor_dim0_stride[31:0] = 128
  g1[6] = 0;                        // stride0 hi, tensor_dim1_stride lo
  g1[7] = 0;
  i32x4 zero4 = {0, 0, 0, 0};       // groups 2/3 unused (2-D tensor)
  i32x8 zero8 = {0, 0, 0, 0, 0, 0, 0, 0};
  __builtin_amdgcn_tensor_load_to_lds(g0, g1, zero4, zero4, zero8, 0);
}

static __device__ __forceinline__ unsigned lds_byte_off(const void* p) {
  // generic pointers into the LDS aperture carry the LDS byte offset in [31:0]
  return (unsigned)(size_t)p;
}

// ---------------------------------------------------------------------------
// Generic f32 WMMA GEMM:  Out[M x Nw] = A[M x K] * W[K x Nw] (+ bias[Nw])
// One wave per 16x16 output tile. grid = (M/16, Nw/16), block = 32.
// ---------------------------------------------------------------------------
__global__ void gemm_f32_wmma(const float* __restrict__ A,
                              const float* __restrict__ W,
                              float* __restrict__ Out,
                              int M, int K, int Nw,
                              const float* __restrict__ bias) {
  const int m0   = blockIdx.x * 16;
  const int n0   = blockIdx.y * 16;
  const int lane = threadIdx.x & 31;
  const int half = lane >> 4;
  const int l16  = lane & 15;
  v8f acc = {};
  const float* Arow = A + (size_t)(m0 + l16) * K;
  for (int k = 0; k < K; k += 4) {
    const int ka = k + half * 2;
    v2f a, b;
    a.x = Arow[ka];
    a.y = Arow[ka + 1];
    const float* Wp = W + (size_t)ka * Nw + n0 + l16;
    b.x = Wp[0];
    b.y = Wp[Nw];
    acc = wmma4(a, b, acc);
  }
#pragma unroll
  for (int r = 0; r < 8; ++r) {
    const int m = m0 + r + half * 8;
    float v = acc[r];
    if (bias) v += bias[n0 + l16];
    Out[(size_t)m * Nw + n0 + l16] = v;
  }
}

// ---------------------------------------------------------------------------
// Apply frames:  g* = R @ pt + t   for gq(48 pts), gk(48 pts), gv(96 pts).
// grid = B*N tokens, block = 96.
// ---------------------------------------------------------------------------
__global__ void frames_kernel(const float* __restrict__ Qf,
                              const float* __restrict__ Kf,
                              const float* __restrict__ Vf,
                              const float* __restrict__ rot,
                              const float* __restrict__ trans,
                              float* __restrict__ gq,
                              float* __restrict__ gk,
                              float* __restrict__ gv) {
  const int tok = blockIdx.x;
  const int t   = threadIdx.x;          // 0..95
  __shared__ float R[9], T[3];
  if (t < 9) R[t] = rot[(size_t)tok * 9 + t];
  if (t < 3) T[t] = trans[(size_t)tok * 3 + t];
  __syncthreads();
  if (t < 48) {
    const float* p = Qf + (size_t)tok * 144 + t * 3;
    float x = p[0], y = p[1], z = p[2];
    float* o = gq + (size_t)tok * 144 + t * 3;
    o[0] = R[0]*x + R[1]*y + R[2]*z + T[0];
    o[1] = R[3]*x + R[4]*y + R[5]*z + T[1];
    o[2] = R[6]*x + R[7]*y + R[8]*z + T[2];
    p = Kf + (size_t)tok * 144 + t * 3;
    x = p[0]; y = p[1]; z = p[2];
    o = gk + (size_t)tok * 144 + t * 3;
    o[0] = R[0]*x + R[1]*y + R[2]*z + T[0];
    o[1] = R[3]*x + R[4]*y + R[5]*z + T[1];
    o[2] = R[6]*x + R[7]*y + R[8]*z + T[2];
  }
  {
    const float* p = Vf + (size_t)tok * 288 + t * 3;
    float x = p[0], y = p[1], z = p[2];
    float* o = gv + (size_t)tok * 288 + t * 3;
    o[0] = R[0]*x + R[1]*y + R[2]*z + T[0];
    o[1] = R[3]*x + R[4]*y + R[5]*z + T[1];
    o[2] = R[6]*x + R[7]*y + R[8]*z + T[2];
  }
}

// ---------------------------------------------------------------------------
// Fused score kernel: scores[b,i,j,h] = WL*( Q.K/4 + bias(z@Wb) + point )
// One wave per (16i x 16j) tile, all 12 heads. The per-i 16x128 z tile is a
// fully contiguous 8 KB block: DMA'd to LDS by TDM, double-buffered, with the
// first DMA issued *before* the head loop so it hides behind ~84 WMMAs.
// grid = (N/16, N/16, B), block = 32.
// ---------------------------------------------------------------------------
__global__ void scores_kernel(const float* __restrict__ Q,
                              const float* __restrict__ Kt,
                              const float* __restrict__ gq,
                              const float* __restrict__ gk,
                              const float* __restrict__ z,
                              const float* __restrict__ Wb,
                              const float* __restrict__ head_w,
                              float* __restrict__ scores) {
  __shared__ float sc[16 * 16 * HH];     // 12 KB composite tile
  __shared__ float zb[2][16 * CZ];       // 2 x 8 KB double-buffered z tiles
  __shared__ float sqq[16], sqk[16];
  const int b    = blockIdx.z;
  const int i0   = blockIdx.x * 16;
  const int j0   = blockIdx.y * 16;
  const int lane = threadIdx.x & 31;
  const int half = lane >> 4;
  const int l16  = lane & 15;

  // kick off DMA of z tile for i = 0 (overlaps with rep/point WMMA below)
  const float* ztile0 = z + (((size_t)(b * NN + i0)) * NN + j0) * CZ;
  tdm_load_z_tile(lds_byte_off(&zb[0][0]), ztile0, 16);

  for (int h = 0; h < HH; ++h) {
    // per-head squared norms of transformed points (12 comps each)
    {
      const float* g = (half == 0)
          ? gq + ((size_t)(b * NN + i0 + l16)) * 144 + h * 12
          : gk + ((size_t)(b * NN + j0 + l16)) * 144 + h * 12;
      float s = 0.f;
#pragma unroll
      for (int q = 0; q < 12; ++q) s += g[q] * g[q];
      if (half == 0) sqq[l16] = s; else sqk[l16] = s;
    }
    __syncthreads();

    // rep:  Q_i (16x16) @ K_j^T  -> K dim = C = 16
    v8f qk = {};
#pragma unroll
    for (int kk = 0; kk < CC; kk += 4) {
      const int ka = kk + half * 2;
      v2f a, bb;
      const float* ap = Q  + ((size_t)(b * NN + i0 + l16)) * (HH*CC) + h * CC + ka;
      a.x = ap[0]; a.y = ap[1];
      const float* bp = Kt + ((size_t)(b * NN + j0 + l16)) * (HH*CC) + h * CC + ka;
      bb.x = bp[0]; bb.y = bp[1];
      qk = wmma4(a, bb, qk);
    }
    // point cross: gq_i . gk_j over 12 comps
    v8f cr = {};
#pragma unroll
    for (int kk = 0; kk < 12; kk += 4) {
      const int ka = kk + half * 2;
      v2f a, bb;
      const float* ap = gq + ((size_t)(b * NN + i0 + l16)) * 144 + h * 12 + ka;
      a.x = ap[0]; a.y = ap[1];
      const float* bp = gk + ((size_t)(b * NN + j0 + l16)) * 144 + h * 12 + ka;
      bb.x = bp[0]; bb.y = bp[1];
      cr = wmma4(a, bb, cr);
    }
    const float gamma  = log1pf(expf(head_w[h]));      // softplus
    const float pscale = -0.5f * gamma * WC_;
#pragma unroll
    for (int r = 0; r < 8; ++r) {
      const int i = r + half * 8;
      const int j = l16;
      const float val = qk[r] * 0.25f +
                        pscale * (sqq[i] + sqk[j] - 2.0f * cr[r]);
      sc[(i * 16 + j) * HH + h] = val;
    }
    __syncthreads();
  }

  // fused bias: for each i, Bias[j,h] = z[b,i,j0+j,:] @ Wb  (K = 128),
  // z tiles arrive via TDM, double-buffered.
  for (int i = 0; i < 16; ++i) {
    if (i + 1 < 16) {
      const float* ztn = z + (((size_t)(b * NN + i0 + i + 1)) * NN + j0) * CZ;
      tdm_load_z_tile(lds_byte_off(&zb[(i + 1) & 1][0]), ztn, 16);
      __builtin_amdgcn_s_wait_tensorcnt(1);   // tile i complete (in-order)
    } else {
      __builtin_amdgcn_s_wait_tensorcnt(0);
    }
    const float* zl = &zb[i & 1][0];
    v8f bias = {};
#pragma unroll 4
    for (int kk = 0; kk < CZ; kk += 4) {
      const int ka = kk + half * 2;
      v2f a, bb;
      a.x = zl[l16 * CZ + ka];                          // A[m=j][k=d] from LDS
      a.y = zl[l16 * CZ + ka + 1];
      bb.x = (l16 < HH) ? Wb[ka * HH + l16] : 0.0f;     // B[k=d][n=h]
      bb.y = (l16 < HH) ? Wb[(ka + 1) * HH + l16] : 0.0f;
      bias = wmma4(a, bb, bias);
    }
#pragma unroll
    for (int r = 0; r < 8; ++r) {
      const int j = r + half * 8;
      const int h = l16;
      if (h < HH) sc[(i * 16 + j) * HH + h] += bias[r];
    }
  }
  __syncthreads();

  // scale by WL and flush tile to global
  for (int e = lane; e < 16 * 16 * HH; e += 32) {
    const int h = e % HH;
    const int j = (e / HH) & 15;
    const int i = e / (HH * 16);
    scores[(((size_t)(b * NN + i0 + i)) * NN + (j0 + j)) * HH + h] = WL_ * sc[e];
  }
}

// ---------------------------------------------------------------------------
// Softmax over j (length 512) per (b,i,h) row. One wave32 per row.
// grid = B*N*H/8, block = 256 (8 waves).
// ---------------------------------------------------------------------------
__global__ void softmax_kernel(float* __restrict__ scores) {
  const int wid  = blockIdx.x * (blockDim.x >> 5) + (threadIdx.x >> 5);
  const int lane = threadIdx.x & 31;
  const int h  = wid % HH;
  const int bi = wid / HH;                       // b*N + i
  float* row = scores + (size_t)bi * NN * HH + h;
  float vals[NN / 32];
  float mx = -3.0e38f;
#pragma unroll
  for (int t = 0; t < NN / 32; ++t) {
    vals[t] = row[(size_t)(t * 32 + lane) * HH];
    mx = fmaxf(mx, vals[t]);
  }
#pragma unroll
  for (int off = 16; off; off >>= 1) mx = fmaxf(mx, __shfl_xor(mx, off, 32));
  float sum = 0.f;
#pragma unroll
  for (int t = 0; t < NN / 32; ++t) { vals[t] = expf(vals[t] - mx); sum += vals[t]; }
#pragma unroll
  for (int off = 16; off; off >>= 1) sum += __shfl_xor(sum, off, 32);
  const float inv = 1.0f / sum;
#pragma unroll
  for (int t = 0; t < NN / 32; ++t)
    row[(size_t)(t * 32 + lane) * HH] = vals[t] * inv;
}

// ---------------------------------------------------------------------------
// out_pair[b,i,h,d] = sum_j attn[b,i,j,h] * z[b,i,j,d]
// Per token: GEMM attn^T(12x512) @ z_i(512x128). z rows (contiguous 16 KB
// chunks of 32x128 f32) are DMA'd to LDS by TDM (wave 0 issues, all 8 waves
// consume via ds_load), double-buffered. Second and last full read of z.
// grid = B*N, block = 256.
// ---------------------------------------------------------------------------
__global__ void out_pair_kernel(const float* __restrict__ attn,
                                const float* __restrict__ z,
                                float* __restrict__ out_pair) {
  __shared__ float zbuf[2][32 * CZ];               // 2 x 16 KB
  const int bi   = blockIdx.x;
  const int w    = threadIdx.x >> 5;
  const int lane = threadIdx.x & 31;
  const int half = lane >> 4;
  const int l16  = lane & 15;
  const int n0   = w * 16;
  const float* arow = attn + (size_t)bi * NN * HH; // [j][h]
  const float* zrow = z    + (size_t)bi * NN * CZ; // [j][d]
  const int CHR = 32, NCH = NN / CHR;              // 16 chunks

  if (w == 0)
    tdm_load_z_tile(lds_byte_off(&zbuf[0][0]), zrow, CHR);

  v8f acc = {};
  for (int c = 0; c < NCH; ++c) {
    if (w == 0) {
      if (c + 1 < NCH) {
        tdm_load_z_tile(lds_byte_off(&zbuf[(c + 1) & 1][0]),
                        zrow + (size_t)(c + 1) * CHR * CZ, CHR);
        __builtin_amdgcn_s_wait_tensorcnt(1);      // chunk c done (in-order)
      } else {
        __builtin_amdgcn_s_wait_tensorcnt(0);
      }
    }
    __syncthreads();
    const float* zl = &zbuf[c & 1][0];
#pragma unroll 4
    for (int kk = 0; kk < CHR; kk += 4) {
      const int jl = kk + half * 2;                // row within chunk
      const int j  = c * CHR + jl;                 // global j
      v2f a, bb;
      a.x = (l16 < HH) ? arow[(size_t)j * HH + l16] : 0.0f;        // A[m=h][k=j]
      a.y = (l16 < HH) ? arow[(size_t)(j + 1) * HH + l16] : 0.0f;
      bb.x = zl[jl * CZ + n0 + l16];                                // B from LDS
      bb.y = zl[(jl + 1) * CZ + n0 + l16];
      acc = wmma4(a, bb, acc);
    }
    __syncthreads();
  }
#pragma unroll
  for (int r = 0; r < 8; ++r) {
    const int h = r + half * 8;
    if (h < HH)
      out_pair[((size_t)bi * HH + h) * CZ + n0 + l16] = acc[r];
  }
}

// ---------------------------------------------------------------------------
// out_rep[b,i,h,:] = attn @ V_h   and  tmp[b,i,h,q,o] = attn @ gv_h
// grid = (N/16, H, B), block = 96 (wave0: V 16 cols, wave1/2: gv 24 cols).
// ---------------------------------------------------------------------------
__global__ void out_rv_kernel(const float* __restrict__ attn,
                              const float* __restrict__ V,
                              const float* __restrict__ gv,
                              float* __restrict__ out_rep,
                              float* __restrict__ tmp) {
  const int i0   = blockIdx.x * 16;
  const int h    = blockIdx.y;
  const int b    = blockIdx.z;
  const int w    = threadIdx.x >> 5;
  const int lane = threadIdx.x & 31;
  const int half = lane >> 4;
  const int l16  = lane & 15;
  v8f acc = {};
  const float* abase = attn + ((size_t)(b * NN + i0 + l16)) * NN * HH + h;
  const int c = (w - 1) * 16 + l16;                // gv column (jobs 1,2)
  for (int kk = 0; kk < NN; kk += 4) {
    const int j = kk + half * 2;
    v2f a, bb;
    a.x = abase[(size_t)j * HH];                   // A[m=i][k=j]
    a.y = abase[(size_t)(j + 1) * HH];
    if (w == 0) {
      const float* vb = V + (size_t)(b * NN) * (HH*CC) + h * CC + l16;
      bb.x = vb[(size_t)j * (HH*CC)];
      bb.y = vb[(size_t)(j + 1) * (HH*CC)];
    } else {
      const float* gb = gv + (size_t)(b * NN) * (HH*PVn*3) + h * (PVn*3);
      bb.x = (c < PVn*3) ? gb[(size_t)j * (HH*PVn*3) + c] : 0.0f;
      bb.y = (c < PVn*3) ? gb[(size_t)(j + 1) * (HH*PVn*3) + c] : 0.0f;
    }
    acc = wmma4(a, bb, acc);
  }
#pragma unroll
  for (int r = 0; r < 8; ++r) {
    const int i = i0 + r + half * 8;
    if (w == 0) {
      out_rep[((size_t)(b * NN + i)) * (HH*CC) + h * CC + l16] = acc[r];
    } else if (c < PVn*3) {
      tmp[((size_t)(b * NN + i)) * (HH*PVn*3) + h * (PVn*3) + c] = acc[r];
    }
  }
}

// ---------------------------------------------------------------------------
// Epilogue: out_vals = R^T (tmp - t), norms, assemble U[b,n,2112].
// grid = B*N, block = 96 (one thread per (h,q) point).
// ---------------------------------------------------------------------------
__global__ void epilogue_kernel(const float* __restrict__ out_rep,
                                const float* __restrict__ tmp,
                                const float* __restrict__ out_pair,
                                const float* __restrict__ rot,
                                const float* __restrict__ trans,
                                float* __restrict__ U) {
  const int tok = blockIdx.x;
  const int t   = threadIdx.x;          // 0..95 = h*8+q
  __shared__ float R[9], T[3];
  if (t < 9) R[t] = rot[(size_t)tok * 9 + t];
  if (t < 3) T[t] = trans[(size_t)tok * 3 + t];
  __syncthreads();
  float* u = U + (size_t)tok * OUTIN;
  {
    const float* p = tmp + (size_t)tok * 288 + t * 3;
    const float x = p[0] - T[0], y = p[1] - T[1], zz = p[2] - T[2];
    const float o0 = R[0]*x + R[3]*y + R[6]*zz;   // R^T rows
    const float o1 = R[1]*x + R[4]*y + R[7]*zz;
    const float o2 = R[2]*x + R[5]*y + R[8]*zz;
    u[192 + t*3 + 0] = o0;
    u[192 + t*3 + 1] = o1;
    u[192 + t*3 + 2] = o2;
    u[480 + t] = sqrtf(o0*o0 + o1*o1 + o2*o2 + 1e-8f);
  }
  for (int e = t; e < 192;  e += 96) u[e]       = out_rep[(size_t)tok * 192 + e];
  for (int e = t; e < 1536; e += 96) u[576 + e] = out_pair[(size_t)tok * 1536 + e];
}

// ---------------------------------------------------------------------------
// Host-side launch
// ---------------------------------------------------------------------------
extern "C" void kernel_launch(void* const* d_in, const int* in_sizes, int n_in,
                              void* d_out, int out_size, void* d_ws, size_t ws_size,
                              hipStream_t stream) {
  const float* s     = (const float*)d_in[0];
  const float* z     = (const float*)d_in[1];
  const float* rot   = (const float*)d_in[2];
  const float* trans = (const float*)d_in[3];
  const float* Wq    = (const float*)d_in[4];
  const float* Wk    = (const float*)d_in[5];
  const float* Wv    = (const float*)d_in[6];
  const float* Wqf   = (const float*)d_in[7];
  const float* Wkf   = (const float*)d_in[8];
  const float* Wvf   = (const float*)d_in[9];
  const float* Wb    = (const float*)d_in[10];
  const float* hw    = (const float*)d_in[11];
  const float* Wo    = (const float*)d_in[12];
  const float* bo    = (const float*)d_in[13];
  float* out = (float*)d_out;
  float* ws  = (float*)d_ws;

  const int M = BB * NN;                  // 1024 tokens
  // workspace layout (float offsets)
  float* Q      = ws;                     // 196608
  float* Kbuf   = ws + 196608;            // 196608
  float* V      = ws + 393216;            // 196608
  float* Qf     = ws + 589824;            // 147456
  float* Kf     = ws + 737280;            // 147456
  float* Vf     = ws + 884736;            // 294912
  float* gq     = ws + 1179648;           // 147456
  float* gk     = ws + 1327104;           // 147456
  float* gv     = ws + 1474560;           // 294912
  float* scores = ws + 1769472;           // 6291456 (25 MB) -> attn in place
  float* orep   = ws + 8060928;           // 196608
  float* tmp    = ws + 8257536;           // 294912
  float* opair  = ws + 8552448;           // 1572864
  float* U      = ws + 10125312;          // 2162688  (end = 12288000 floats)

  // 1) projections: s(1024x384) @ W  -> all WMMA
  gemm_f32_wmma<<<dim3(M/16, 192/16), 32, 0, stream>>>(s, Wq,  Q,    M, CS, 192, nullptr);
  gemm_f32_wmma<<<dim3(M/16, 192/16), 32, 0, stream>>>(s, Wk,  Kbuf, M, CS, 192, nullptr);
  gemm_f32_wmma<<<dim3(M/16, 192/16), 32, 0, stream>>>(s, Wv,  V,    M, CS, 192, nullptr);
  gemm_f32_wmma<<<dim3(M/16, 144/16), 32, 0, stream>>>(s, Wqf, Qf,   M, CS, 144, nullptr);
  gemm_f32_wmma<<<dim3(M/16, 144/16), 32, 0, stream>>>(s, Wkf, Kf,   M, CS, 144, nullptr);
  gemm_f32_wmma<<<dim3(M/16, 288/16), 32, 0, stream>>>(s, Wvf, Vf,   M, CS, 288, nullptr);

  // 2) frame transforms
  frames_kernel<<<M, 96, 0, stream>>>(Qf, Kf, Vf, rot, trans, gq, gk, gv);

  // 3) fused scores (rep + point + z@Wb bias), z tiles via TDM double-buffer
  scores_kernel<<<dim3(NN/16, NN/16, BB), 32, 0, stream>>>(
      Q, Kbuf, gq, gk, z, Wb, hw, scores);

  // 4) softmax over j (wave per row)
  softmax_kernel<<<(M * HH) / 8, 256, 0, stream>>>(scores);

  // 5) out_pair = attn^T @ z  (TDM-staged, second and last read of z)
  out_pair_kernel<<<M, 256, 0, stream>>>(scores, z, opair);

  // 6) out_rep = attn @ V, tmp = attn @ gv
  out_rv_kernel<<<dim3(NN/16, HH, BB), 96, 0, stream>>>(scores, V, gv, orep, tmp);

  // 7) inverse frames, norms, concat into U
  epilogue_kernel<<<M, 96, 0, stream>>>(orep, tmp, opair, rot, trans, U);

  // 8) final GEMM: out = U(1024x2112) @ Wo(2112x384) + bo
  gemm_f32_wmma<<<dim3(M/16, CS/16), 32, 0, stream>>>(U, Wo, out, M, OUTIN, CS, bo);
}